// HashEmbedder3D_42262478193062
// MI455X (gfx1250) — compile-verified
//
#include <hip/hip_runtime.h>
#include <cmath>
#include <cstdint>

typedef float v2f __attribute__((ext_vector_type(2)));

#define NLVL    16
#define BLOCK   512
#define LDS_CAP 13056   /* float2 entries -> 104,448 bytes of static LDS (levels 0..5) */
#define HASH_MASK 524287u

struct LevelP {
  float gs0, gs1, gs2;     // grid size = IMG / res  (float32, matches reference)
  float inv0, inv1, inv2;  // host-precomputed 1.0f/gs (uniform per level)
  int   s1, s2;            // linear strides for one-to-one levels (res1*res2, res2)
  int   one2one;
  int   lds_off;           // entry offset into LDS staging buffer (staged levels)
  int   nent;              // number of float2 entries in this level's table
};

struct KParams {
  LevelP     L[NLVL];
  const v2f* tab[NLVL];
};

template <int NSTAGED>
__global__ __launch_bounds__(BLOCK)
void hashenc3d_kernel(const float* __restrict__ x, float* __restrict__ out,
                      KParams P, int B) {
  __shared__ v2f stab[LDS_CAP];
  const int tid = threadIdx.x;

  // ---- Stage small-level tables into LDS via CDNA5 async global->LDS copies.
  // LDS address is derived from the real &stab[...] pointer (low 32 bits of a
  // generic LDS pointer == LDS byte offset) so stab escapes; the "memory"
  // clobber keeps the ds_load consumption path alive.
#pragma unroll
  for (int l = 0; l < NSTAGED; ++l) {
    const v2f* src  = P.tab[l];
    const int  n    = P.L[l].nent;
    const int  loff = P.L[l].lds_off;
    for (int e = tid; e < n; e += BLOCK) {
      v2f* dst = &stab[loff + e];
      unsigned laddr = (unsigned)(uintptr_t)dst;
      unsigned long long gaddr = (unsigned long long)(const void*)(src + e);
      asm volatile("global_load_async_to_lds_b64 %0, %1, off"
                   :: "v"(laddr), "v"(gaddr) : "memory");
    }
  }
  if (NSTAGED > 0) {
    asm volatile("s_wait_asynccnt 0x0" ::: "memory");
    __syncthreads();
  }

  int b = blockIdx.x * BLOCK + tid;
  if (b >= B) return;

  const float x0 = x[3 * b + 0];
  const float x1 = x[3 * b + 1];
  const float x2 = x[3 * b + 2];

#pragma unroll
  for (int l = 0; l < NLVL; ++l) {
    const LevelP p = P.L[l];

    // bl = floor(x/gs); gmin = bl*gs; w = (x - gmin)/gs
    // Divides replaced by multiplies with a host-precomputed reciprocal of
    // the uniform gs (~1 ulp difference; interpolation is continuous across
    // cell boundaries so boundary flips are numerically harmless).
    const float t0 = x0 * p.inv0, t1 = x1 * p.inv1, t2 = x2 * p.inv2;
    const float f0 = floorf(t0), f1 = floorf(t1), f2 = floorf(t2);
    const int   i0 = (int)f0,    i1 = (int)f1,    i2 = (int)f2;
    const float w0 = (x0 - f0 * p.gs0) * p.inv0;
    const float w1 = (x1 - f1 * p.gs1) * p.inv1;
    const float w2 = (x2 - f2 * p.gs2) * p.inv2;

    // corner flat order: 4*i + 2*j + k
    unsigned idx[8];
    if (p.one2one) {
      const int base = i0 * p.s1 + i1 * p.s2 + i2;
      idx[0] = (unsigned)(base);
      idx[1] = (unsigned)(base + 1);
      idx[2] = (unsigned)(base + p.s2);
      idx[3] = (unsigned)(base + p.s2 + 1);
      idx[4] = (unsigned)(base + p.s1);
      idx[5] = (unsigned)(base + p.s1 + 1);
      idx[6] = (unsigned)(base + p.s1 + p.s2);
      idx[7] = (unsigned)(base + p.s1 + p.s2 + 1);
    } else {
      const unsigned u0 = (unsigned)i0, u1 = (unsigned)i1, u2 = (unsigned)i2;
      const unsigned a0 = u0,               a1 = u0 + 1u;
      const unsigned b0 = u1 * 2654435761u, b1 = (u1 + 1u) * 2654435761u;
      const unsigned c0 = u2 * 805459861u,  c1 = (u2 + 1u) * 805459861u;
      idx[0] = (a0 ^ b0 ^ c0) & HASH_MASK;
      idx[1] = (a0 ^ b0 ^ c1) & HASH_MASK;
      idx[2] = (a0 ^ b1 ^ c0) & HASH_MASK;
      idx[3] = (a0 ^ b1 ^ c1) & HASH_MASK;
      idx[4] = (a1 ^ b0 ^ c0) & HASH_MASK;
      idx[5] = (a1 ^ b0 ^ c1) & HASH_MASK;
      idx[6] = (a1 ^ b1 ^ c0) & HASH_MASK;
      idx[7] = (a1 ^ b1 ^ c1) & HASH_MASK;
    }

    // Gather all 8 corners up-front so loads batch and overlap latency.
    v2f e[8];
    if (l < NSTAGED) {
#pragma unroll
      for (int c = 0; c < 8; ++c) e[c] = stab[p.lds_off + (int)idx[c]];
    } else {
      const v2f* __restrict__ t = P.tab[l];
#pragma unroll
      for (int c = 0; c < 8; ++c) e[c] = t[idx[c]];
    }

    const float om0 = 1.0f - w0, om1 = 1.0f - w1, om2 = 1.0f - w2;
    const v2f c00 = e[0] * om0 + e[4] * w0;   // lerp over i
    const v2f c01 = e[1] * om0 + e[5] * w0;
    const v2f c10 = e[2] * om0 + e[6] * w0;
    const v2f c11 = e[3] * om0 + e[7] * w0;
    const v2f d0  = c00 * om1 + c10 * w1;     // lerp over j
    const v2f d1  = c01 * om1 + c11 * w1;
    const v2f o   = d0 * om2 + d1 * w2;       // lerp over k

    __builtin_nontemporal_store(o, (v2f*)(out + 32 * b + 2 * l));
  }
}

extern "C" void kernel_launch(void* const* d_in, const int* in_sizes, int n_in,
                              void* d_out, int out_size, void* d_ws, size_t ws_size,
                              hipStream_t stream) {
  const float* x   = (const float*)d_in[0];
  float*       out = (float*)d_out;

  // Replicate the reference's float32 constant math on the host.
  const float IMGC[3]  = {75.0f, 256.0f, 256.0f};
  const float BASEc[3] = {3.0f, 8.0f, 8.0f};
  float bg[3];
  for (int d = 0; d < 3; ++d)
    bg[d] = expf((logf(IMGC[d]) - logf(BASEc[d])) / 15.0f);

  KParams P;
  int staged_entries = 0, n_staged = 0;
  for (int l = 0; l < NLVL; ++l) {
    float res[3];
    for (int d = 0; d < 3; ++d)
      res[d] = floorf(BASEc[d] * powf(bg[d], (float)l));

    LevelP& L = P.L[l];
    L.gs0  = IMGC[0] / res[0];
    L.gs1  = IMGC[1] / res[1];
    L.gs2  = IMGC[2] / res[2];
    L.inv0 = 1.0f / L.gs0;
    L.inv1 = 1.0f / L.gs1;
    L.inv2 = 1.0f / L.gs2;
    L.s1   = (int)(res[1] * res[2]);
    L.s2   = (int)res[2];

    const int entries = in_sizes[1 + l] / 2;   // ground truth table size
    L.nent    = entries;
    L.one2one = (entries != (1 << 19)) ? 1 : 0;

    // Greedily stage a contiguous prefix of levels into LDS.
    if (n_staged == l && staged_entries + entries <= LDS_CAP) {
      L.lds_off = staged_entries;
      staged_entries += entries;
      n_staged++;
    } else {
      L.lds_off = 0;
    }
    P.tab[l] = (const v2f*)d_in[1 + l];
  }

  const int B    = in_sizes[0] / 3;
  const int grid = (B + BLOCK - 1) / BLOCK;

  // Only two configurations can matter here: the expected 6-level staged
  // prefix (this problem's table sizes), or no staging as a safe fallback.
  if (n_staged >= 6) {
    hipLaunchKernelGGL(hashenc3d_kernel<6>, dim3(grid), dim3(BLOCK), 0, stream, x, out, P, B);
  } else {
    hipLaunchKernelGGL(hashenc3d_kernel<0>, dim3(grid), dim3(BLOCK), 0, stream, x, out, P, B);
  }
}